// GNN_23424751632407
// MI455X (gfx1250) — compile-verified
//
#include <hip/hip_runtime.h>

typedef __attribute__((ext_vector_type(2))) float v2f;
typedef __attribute__((ext_vector_type(8))) float v8f;

#define N_NODES  100000
#define N_EDGES  800000
#define HIDF     128
#define EMBF     64
#define OUTF     8
#define N_GRAPHS 64
#define N_MTILES (N_NODES / 16)   // 6250, exact
#define WPITCH   132              // LDS row pitch (floats): 16B-aligned rows, conflict-free b64 reads
#define SMEM_BYTES (2 * HIDF * WPITCH * (int)sizeof(float))   // 135168 B < 320KB/WGP

// ---------------------------------------------------------------------------
// zero fill
// ---------------------------------------------------------------------------
__global__ void gnn_zero_f32(float* __restrict__ p, int n) {
    int i = blockIdx.x * blockDim.x + threadIdx.x;
    if (i < n) p[i] = 0.0f;
}

// ---------------------------------------------------------------------------
// scatter-add h[src] into agg[dst]; one wave (32 lanes) per edge, 4 floats/lane.
// lane 0 also counts the in-degree of dst.
// ---------------------------------------------------------------------------
__global__ void gnn_scatter_accum(const float* __restrict__ h,
                                  const int* __restrict__ src,
                                  const int* __restrict__ dst,
                                  float* __restrict__ agg,
                                  float* __restrict__ deg) {
    int tid  = blockIdx.x * blockDim.x + threadIdx.x;
    int e    = tid >> 5;
    if (e >= N_EDGES) return;
    int lane = tid & 31;
    int s = src[e];
    int d = dst[e];
    const float4 v = *(const float4*)(h + (size_t)s * HIDF + lane * 4);
    float* a = agg + (size_t)d * HIDF + lane * 4;
    atomicAdd(a + 0, v.x);
    atomicAdd(a + 1, v.y);
    atomicAdd(a + 2, v.z);
    atomicAdd(a + 3, v.w);
    if (lane == 0) atomicAdd(deg + d, 1.0f);
}

// deg[i] -> 1 / max(deg[i], 1)   (PyG scatter-'mean' clamp)
__global__ void gnn_inv_deg(float* __restrict__ deg, int n) {
    int i = blockIdx.x * blockDim.x + threadIdx.x;
    if (i < n) deg[i] = 1.0f / fmaxf(deg[i], 1.0f);
}

// ---------------------------------------------------------------------------
// Fused SAGE linear:  OUT = relu( (AGG*invdeg) @ Wl^T + bl + H @ Wr^T )
// fp32 WMMA 16x16x4; weights staged in LDS (padded pitch, conflict-free).
// One wave per 16-node row tile, all 8 column tiles (64 f32 accum VGPRs).
// In-place safe with OUT == AGG (a wave reads only the rows it writes).
//
// A layout (16x4 f32): lane l<16 -> M=l, K=0..1 ; lanes 16..31 -> M=l, K=2..3
// B layout (4x16):     VGPR v, lanes 0-15 -> K=v,  N=lane ; lanes 16-31 -> K=v+2
// D layout (16x16):    VGPR v, lanes 0-15 -> M=v,  N=lane ; lanes 16-31 -> M=v+8
// ---------------------------------------------------------------------------
__global__ __launch_bounds__(256)
void gnn_sage_gemm_relu(const float* __restrict__ AGG,
                        const float* __restrict__ H,
                        const float* __restrict__ Wl,
                        const float* __restrict__ bl,
                        const float* __restrict__ Wr,
                        const float* __restrict__ invdeg,
                        float* __restrict__ OUT) {
    extern __shared__ float smem[];          // [2][HIDF][WPITCH]
    float* sWl = smem;
    float* sWr = smem + HIDF * WPITCH;

    // cooperative stage of both weight matrices into LDS (float4 granularity)
    for (int i = threadIdx.x; i < (HIDF * HIDF) / 4; i += blockDim.x) {
        const int row = (i * 4) >> 7;        // /128
        const int col = (i * 4) & 127;
        *(float4*)(sWl + row * WPITCH + col) = *(const float4*)(Wl + i * 4);
        *(float4*)(sWr + row * WPITCH + col) = *(const float4*)(Wr + i * 4);
    }
    __syncthreads();

    const int wave = threadIdx.x >> 5;
    const int lane = threadIdx.x & 31;
    const int half = lane >> 4;              // 0: K={0,1}, 1: K={2,3}
    const int l    = lane & 15;

    const int mt = blockIdx.x * (blockDim.x >> 5) + wave;
    if (mt >= N_MTILES) return;              // wave-uniform: EXEC all-ones below
    const int m0 = mt * 16;

    const int   arow = m0 + l;
    const float inv  = invdeg[arow];
    const float* aRow = AGG + (size_t)arow * HIDF;
    const float* hRow = H   + (size_t)arow * HIDF;

    v8f acc[8] = {};

    #pragma unroll 2
    for (int kk = 0; kk < 32; ++kk) {
        const int k0 = kk * 4 + half * 2;
        v2f av = *(const v2f*)(aRow + k0);
        av.x *= inv;
        av.y *= inv;
        const v2f hv = *(const v2f*)(hRow + k0);

        // batch all B fragments from LDS first, then the WMMA chain
        v2f bfl[8], bfr[8];
        #pragma unroll
        for (int nt = 0; nt < 8; ++nt) {
            const int ncol = nt * 16 + l;
            bfl[nt] = *(const v2f*)(sWl + ncol * WPITCH + k0);
            bfr[nt] = *(const v2f*)(sWr + ncol * WPITCH + k0);
        }
        #pragma unroll
        for (int nt = 0; nt < 8; ++nt) {
            acc[nt] = __builtin_amdgcn_wmma_f32_16x16x4_f32(
                false, av, false, bfl[nt], (short)0, acc[nt], false, false);
            acc[nt] = __builtin_amdgcn_wmma_f32_16x16x4_f32(
                false, hv, false, bfr[nt], (short)0, acc[nt], false, false);
        }
    }

    #pragma unroll
    for (int nt = 0; nt < 8; ++nt) {
        const int   ncol = nt * 16 + l;
        const float bias = bl[ncol];
        #pragma unroll
        for (int v = 0; v < 8; ++v) {
            const int row = m0 + v + half * 8;
            const float val = acc[nt][v] + bias;
            OUT[(size_t)row * HIDF + ncol] = fmaxf(val, 0.0f);
        }
    }
}

// ---------------------------------------------------------------------------
// graph mean pool: one wave per node, 4 floats/lane into pooled[batch[n]]
// ---------------------------------------------------------------------------
__global__ void gnn_pool_accum(const float* __restrict__ h,
                               const int* __restrict__ batch,
                               float* __restrict__ pooled,
                               float* __restrict__ gcnt) {
    int tid  = blockIdx.x * blockDim.x + threadIdx.x;
    int n    = tid >> 5;
    if (n >= N_NODES) return;
    int lane = tid & 31;
    int g = batch[n];
    const float4 v = *(const float4*)(h + (size_t)n * HIDF + lane * 4);
    float* p = pooled + (size_t)g * HIDF + lane * 4;
    atomicAdd(p + 0, v.x);
    atomicAdd(p + 1, v.y);
    atomicAdd(p + 2, v.z);
    atomicAdd(p + 3, v.w);
    if (lane == 0) atomicAdd(gcnt + g, 1.0f);
}

// out[g][o] = (pooled[g]/cnt) . Wlin[o][0:128] + embed[g] . Wlin[o][128:192] + blin[o]
__global__ void gnn_final_linear(const float* __restrict__ pooled,
                                 const float* __restrict__ gcnt,
                                 const float* __restrict__ embed,
                                 const float* __restrict__ Wlin,
                                 const float* __restrict__ blin,
                                 float* __restrict__ out) {
    int t = threadIdx.x;                  // 512 threads: (graph, out-feature)
    int g = t >> 3;
    int o = t & 7;
    const float invc = 1.0f / fmaxf(gcnt[g], 1.0f);
    const float* w = Wlin + (size_t)o * (HIDF + EMBF);
    float s = blin[o];
    #pragma unroll 8
    for (int f = 0; f < HIDF; ++f) s += pooled[(size_t)g * HIDF + f] * invc * w[f];
    #pragma unroll 8
    for (int e = 0; e < EMBF; ++e) s += embed[(size_t)g * EMBF + e] * w[HIDF + e];
    out[g * OUTF + o] = s;
}

// ---------------------------------------------------------------------------
extern "C" void kernel_launch(void* const* d_in, const int* in_sizes, int n_in,
                              void* d_out, int out_size, void* d_ws, size_t ws_size,
                              hipStream_t stream) {
    const float* x     = (const float*)d_in[0];
    const int*   eidx  = (const int*)d_in[1];      // [2, N_EDGES] row-major
    const int*   batch = (const int*)d_in[2];
    const float* embed = (const float*)d_in[3];
    const float* W1l   = (const float*)d_in[4];
    const float* b1l   = (const float*)d_in[5];
    const float* W1r   = (const float*)d_in[6];
    const float* W2l   = (const float*)d_in[7];
    const float* b2l   = (const float*)d_in[8];
    const float* W2r   = (const float*)d_in[9];
    const float* Wlin  = (const float*)d_in[10];
    const float* blin  = (const float*)d_in[11];
    float* out = (float*)d_out;

    const int* esrc = eidx;
    const int* edst = eidx + N_EDGES;

    // workspace layout (floats): [bufA NxH][bufB NxH][deg N][pooled GxH][gcnt G]
    float* bufA   = (float*)d_ws;
    float* bufB   = bufA + (size_t)N_NODES * HIDF;
    float* deg    = bufB + (size_t)N_NODES * HIDF;
    float* pooled = deg + N_NODES;
    float* gcnt   = pooled + (size_t)N_GRAPHS * HIDF;

    const int ZT = 256;
    const int nodeF   = N_NODES * HIDF;
    const int edgeThr = N_EDGES * 32;
    const int poolThr = N_NODES * 32;
    const int gemmBlk = 256;                                  // 8 waves/block
    const int gemmGrd = (N_MTILES + 7) / 8;

    // ---- layer 1: agg(x) -> bufA ; h1 = relu(gemm) in-place in bufA ----
    gnn_zero_f32<<<(nodeF + ZT - 1) / ZT, ZT, 0, stream>>>(bufA, nodeF);
    gnn_zero_f32<<<(N_NODES + ZT - 1) / ZT, ZT, 0, stream>>>(deg, N_NODES);
    gnn_scatter_accum<<<(edgeThr + ZT - 1) / ZT, ZT, 0, stream>>>(x, esrc, edst, bufA, deg);
    gnn_inv_deg<<<(N_NODES + ZT - 1) / ZT, ZT, 0, stream>>>(deg, N_NODES);
    gnn_sage_gemm_relu<<<gemmGrd, gemmBlk, SMEM_BYTES, stream>>>(bufA, x, W1l, b1l, W1r, deg, bufA);

    // ---- layer 2: agg(h1) -> bufB ; h2 = relu(gemm) in-place in bufB ----
    gnn_zero_f32<<<(nodeF + ZT - 1) / ZT, ZT, 0, stream>>>(bufB, nodeF);
    gnn_zero_f32<<<(N_NODES + ZT - 1) / ZT, ZT, 0, stream>>>(deg, N_NODES);
    gnn_scatter_accum<<<(edgeThr + ZT - 1) / ZT, ZT, 0, stream>>>(bufA, esrc, edst, bufB, deg);
    gnn_inv_deg<<<(N_NODES + ZT - 1) / ZT, ZT, 0, stream>>>(deg, N_NODES);
    gnn_sage_gemm_relu<<<gemmGrd, gemmBlk, SMEM_BYTES, stream>>>(bufB, bufA, W2l, b2l, W2r, deg, bufB);

    // ---- pool + head ----
    gnn_zero_f32<<<(N_GRAPHS * HIDF + ZT - 1) / ZT, ZT, 0, stream>>>(pooled, N_GRAPHS * HIDF);
    gnn_zero_f32<<<1, ZT, 0, stream>>>(gcnt, N_GRAPHS);
    gnn_pool_accum<<<(poolThr + ZT - 1) / ZT, ZT, 0, stream>>>(bufB, batch, pooled, gcnt);
    gnn_final_linear<<<1, N_GRAPHS * OUTF, 0, stream>>>(pooled, gcnt, embed, Wlin, blin, out);
}